// SpatialEngram_38199439131338
// MI455X (gfx1250) — compile-verified
//
#include <hip/hip_runtime.h>
#include <hip/hip_bf16.h>

// ---------------------------------------------------------------------------
// SpatialEngram on MI455X (gfx1250, wave32)
//
// Roofline: read x (128 MB) + write out (256 MB) + idx scratch (8 MB) at
// 23.3 TB/s => ~17 us floor. GEMM is only 8.6 GFLOP -> WMMA makes it free.
// Pass 1: LDS-tiled 3x3 quantized window sum -> idx (int32, in d_ws).
// Pass 2: table gather (L2-resident, 2.5 MB) + V_WMMA_F32_16X16X4_F32 GEMM.
//         One wave owns ALL 64 output channels for its 16-pixel tile:
//         B (gathered emb^T) is loaded once and feeds 64 WMMAs across 4
//         independent accumulator chains -> 1x (not 4x) L2 gather traffic,
//         proj_w loop-invariant in 128 VGPRs, bias fused into C init,
//         coalesced per-o-plane stores.
// ---------------------------------------------------------------------------

#define BB   4
#define CC   32
#define HH   512
#define WW   512
#define HWSZ (HH * WW)          // 2^18
#define NPAT 10000
#define EE   64
#define OO   64

typedef float v2f __attribute__((ext_vector_type(2)));
typedef float v8f __attribute__((ext_vector_type(8)));

// ---------------------------------------------------------------------------
// Pass 1: per-pixel pattern index.
// Block = 256 threads = one 16x16 output tile. LDS holds the 18x18 quantized
// halo tile for one channel at a time (row stride 20 to dodge bank conflicts).
// ---------------------------------------------------------------------------
__global__ __launch_bounds__(256)
void engram_idx_kernel(const float* __restrict__ x, int* __restrict__ idx) {
    __shared__ int q[18 * 20];

    const int t  = threadIdx.x;
    const int tx = t & 15;
    const int ty = t >> 4;
    const int b  = blockIdx.z;
    const int x0 = blockIdx.x * 16;
    const int y0 = blockIdx.y * 16;

    // Each thread cooperatively loads element t (and t+256 if <324) of the
    // 18x18 halo. Replicate padding == clamp of global coords.
    const int e0  = t;
    const int e1  = t + 256;
    const bool h1 = (e1 < 18 * 18);
    const int ex0 = e0 % 18, ey0 = e0 / 18;
    const int ex1 = e1 % 18, ey1 = e1 / 18;
    const int gx0 = min(max(x0 - 1 + ex0, 0), WW - 1);
    const int gy0 = min(max(y0 - 1 + ey0, 0), HH - 1);
    const int gx1 = min(max(x0 - 1 + ex1, 0), WW - 1);
    const int gy1 = min(max(y0 - 1 + ey1, 0), HH - 1);
    const size_t off0 = (size_t)gy0 * WW + gx0;
    const size_t off1 = (size_t)gy1 * WW + gx1;

    const float* xb = x + (size_t)b * CC * HWSZ;

    float acc = 0.0f;
    for (int c = 0; c < CC; ++c) {
        const float* xc = xb + (size_t)c * HWSZ;
        // trunc-toward-zero quantization: C float->int cast == jnp.trunc
        q[ey0 * 20 + ex0] = (int)(xc[off0] * 100.0f);
        if (h1) q[ey1 * 20 + ex1] = (int)(xc[off1] * 100.0f);
        __syncthreads();

        int s = 0;
#pragma unroll
        for (int di = 0; di < 3; ++di) {
#pragma unroll
            for (int dj = 0; dj < 3; ++dj) {
                s += q[(ty + di) * 20 + (tx + dj)];
            }
        }
        int a = (s < 0) ? -s : s;
        acc += (float)(a % NPAT);
        __syncthreads();
    }

    // mean over 32 channels: sum < 320000 < 2^24 -> exact in f32; /32 exact.
    const int id = (int)(acc * (1.0f / 32.0f));
    idx[((size_t)b * HH + (y0 + ty)) * WW + (x0 + tx)] = id;
}

// ---------------------------------------------------------------------------
// Pass 2: gather + WMMA GEMM.
// Block = 256 threads = 8 waves; each wave independently grid-strides over
// 16-pixel tiles and computes all 64 output channels (4 o-tiles).
// Consecutive tiles land on consecutive waves of a block so the 64 B
// half-wave output writes merge into full 256 B lines in L2.
// ---------------------------------------------------------------------------
__global__ __launch_bounds__(256)
void engram_gemm_kernel(const int* __restrict__ idx,
                        const float* __restrict__ table,
                        const float* __restrict__ pw,
                        const float* __restrict__ pb,
                        float* __restrict__ out,
                        int num_tiles) {
    const int lane = threadIdx.x & 31;
    const int wave = threadIdx.x >> 5;       // 0..7
    const int half = lane >> 4;              // ISA: lanes 16-31 hold K+2 / M+8
    const int lp   = lane & 15;

    // --- A matrices (16x4 per K-step, one per o-tile m): lane lp = row M
    // (= o within tile), half picks the K pair. A[m][kk] covers
    // K = 4*kk + 2*half + {0,1}. Loop-invariant: 128 VGPRs total.
    v2f A[4][16];
    float bias[4][8];
#pragma unroll
    for (int m = 0; m < 4; ++m) {
        const int o_row = m * 16 + lp;
#pragma unroll
        for (int kk = 0; kk < 16; ++kk)
            A[m][kk] = *(const v2f*)(pw + (size_t)o_row * EE + 4 * kk + 2 * half);
        // Bias in D layout: VGPR r, lane l -> o = 16m + r + 8*half.
#pragma unroll
        for (int r = 0; r < 8; ++r)
            bias[m][r] = pb[m * 16 + r + 8 * half];
    }

    const int wid = blockIdx.x * 8 + wave;
    const int nw  = gridDim.x * 8;
    for (int t = wid; t < num_tiles; t += nw) {
        const int p   = t * 16 + lp;         // this lane's pixel (dup per half)
        const int row = idx[p];
        const float* tr = table + (size_t)row * EE;

        // --- B matrix (4x16 per K-step): VGPR j, lane l -> K = j + 2*half,
        // N = lp. float2 at e = 4*kk + 2*half gives exactly {j=0, j=1}.
        // Loaded ONCE per tile, feeds all 64 WMMAs.
        v2f Bm[16];
#pragma unroll
        for (int kk = 0; kk < 16; ++kk)
            Bm[kk] = *(const v2f*)(tr + 4 * kk + 2 * half);

        v8f c[4];
#pragma unroll
        for (int m = 0; m < 4; ++m)
#pragma unroll
            for (int r = 0; r < 8; ++r) c[m][r] = bias[m][r];

        // 4 independent accumulator chains interleaved -> RAW distance 4
        // on the matrix pipe instead of back-to-back.
#pragma unroll
        for (int kk = 0; kk < 16; ++kk)
#pragma unroll
            for (int m = 0; m < 4; ++m)
                c[m] = __builtin_amdgcn_wmma_f32_16x16x4_f32(
                           /*neg_a=*/false, A[m][kk],
                           /*neg_b=*/false, Bm[kk],
                           /*c_mod=*/(short)0, c[m],
                           /*reuse_a=*/false, /*reuse_b=*/false);

        // --- Store: D VGPR r, lane l -> out[b][16m + r + 8*half][hw0 + lp].
        // 16-aligned tiles never cross a row/batch boundary (W, H*W % 16 == 0)
        // so lanes 0-15 / 16-31 each write 64 contiguous bytes.
        const int b  = p >> 18;              // H*W == 2^18
        const int hw = p & (HWSZ - 1);
#pragma unroll
        for (int m = 0; m < 4; ++m) {
            float* ob = out + ((size_t)(b * OO + m * 16 + 8 * half) << 18) + hw;
#pragma unroll
            for (int r = 0; r < 8; ++r)
                ob[(size_t)r << 18] = c[m][r];
        }
    }
}

extern "C" void kernel_launch(void* const* d_in, const int* in_sizes, int n_in,
                              void* d_out, int out_size, void* d_ws, size_t ws_size,
                              hipStream_t stream) {
    const float* x     = (const float*)d_in[0];   // (4,32,512,512) f32
    const float* table = (const float*)d_in[1];   // (10000,64) f32
    const float* pw    = (const float*)d_in[2];   // (64,64) f32
    const float* pb    = (const float*)d_in[3];   // (64,) f32
    float* out = (float*)d_out;                   // (4,64,512,512) f32
    int* idx = (int*)d_ws;                        // 4*512*512*4 = 4 MB scratch

    dim3 g1(WW / 16, HH / 16, BB);
    engram_idx_kernel<<<g1, 256, 0, stream>>>(x, idx);

    const int num_tiles = (BB * HWSZ) / 16;       // 65536
    engram_gemm_kernel<<<1024, 256, 0, stream>>>(idx, table, pw, pb, out, num_tiles);
}